// OutputBlock_83665962926265
// MI455X (gfx1250) — compile-verified
//
#include <hip/hip_runtime.h>

#define N_EDGES    320000
#define N_PART     10000
#define EMBED      256
#define N_RBF      16
#define OUTE       512

typedef __attribute__((ext_vector_type(2))) float v2f;
typedef __attribute__((ext_vector_type(8))) float v8f;

// ---------------------------------------------------------------- zero fill
__global__ void zero_f32(float* __restrict__ p, int n) {
  int i = blockIdx.x * blockDim.x + threadIdx.x;
  if (i < n) p[i] = 0.0f;
}

// ------------------------------------------- edge gate + scatter (fp32 atomics)
// block = 256 threads (one per embed dim), 16 edges per block.
__global__ __launch_bounds__(256) void edge_gate_scatter(
    const float* __restrict__ messages,   // [N_EDGES, EMBED]
    const float* __restrict__ rbf,        // [N_EDGES, N_RBF]
    const int*   __restrict__ idx,        // [N_EDGES]
    const float* __restrict__ W_rbf,      // [N_RBF, EMBED]
    float* __restrict__ summed,           // [N_PART, EMBED]
    int n_edges) {
  __shared__ float w[N_RBF * EMBED];      // 16 KB
  __shared__ float srbf[16][N_RBF];
  __shared__ int   sidx[16];

  for (int i = threadIdx.x; i < N_RBF * EMBED; i += 256) w[i] = W_rbf[i];

  const int e0 = blockIdx.x * 16;
  {
    // 16 edges x 16 rbf values = 256 coalesced loads
    int ei = threadIdx.x >> 4;
    int k  = threadIdx.x & 15;
    int e  = e0 + ei;
    srbf[ei][k] = (e < n_edges) ? rbf[(size_t)e * N_RBF + k] : 0.0f;
    if (threadIdx.x < 16) {
      int ee = e0 + threadIdx.x;
      sidx[threadIdx.x] = (ee < n_edges) ? idx[ee] : 0;
    }
  }
  __syncthreads();

  const int d = threadIdx.x;              // embed dim for this thread
  for (int ei = 0; ei < 16; ++ei) {
    int e = e0 + ei;
    if (e >= n_edges) break;
    float t = 0.0f;
#pragma unroll
    for (int k = 0; k < N_RBF; ++k)
      t = __builtin_fmaf(srbf[ei][k], w[k * EMBED + d], t);
    float g = messages[(size_t)e * EMBED + d] * t;
    atomicAdd(&summed[(size_t)sidx[ei] * EMBED + d], g);   // L2-resident fp32 atomic
  }
}

// -------------------------------------------------- fp32 WMMA GEMM (+bias+swish)
// C[M,N] = act(A[M,K] @ B[K,N] + bias). block = 8 waves; block covers 16 rows
// x 512 cols; each wave owns a 16x64 strip (4 accumulator tiles).
template <int ACT>
__global__ __launch_bounds__(256) void gemm_wmma_f32(
    const float* __restrict__ A, const float* __restrict__ B,
    const float* __restrict__ bias, float* __restrict__ C,
    int M, int K, int N) {
  const int wave = threadIdx.x >> 5;
  const int lane = threadIdx.x & 31;
  const int half = lane >> 4;             // 0: K lo pair, 1: K hi pair
  const int r    = lane & 15;
  const int m0   = blockIdx.x * 16;
  const int n0   = wave * 64;

  const v8f zero = {0.f, 0.f, 0.f, 0.f, 0.f, 0.f, 0.f, 0.f};
  v8f acc[4] = {zero, zero, zero, zero};

  const float* arow = A + (size_t)(m0 + r) * K;

  for (int k = 0; k < K; k += 4) {
    const int kk = k + 2 * half;                      // even -> 8B aligned
    // A 16x4 fragment: lane r(+16*half) holds row m0+r, K = kk, kk+1
    v2f a = *(const v2f*)(arow + kk);
#pragma unroll
    for (int t = 0; t < 4; ++t) {
      const int col = n0 + t * 16 + r;
      // B 4x16 fragment: lane holds col, rows kk / kk+1 (coalesced over lanes)
      v2f b;
      b.x = B[(size_t)kk * N + col];
      b.y = B[(size_t)(kk + 1) * N + col];
      acc[t] = __builtin_amdgcn_wmma_f32_16x16x4_f32(
          false, a, false, b, (short)0, acc[t], false, false);
    }
  }

  // C/D layout: VGPR i -> row m0 + i + 8*half, col n0 + t*16 + r
#pragma unroll
  for (int t = 0; t < 4; ++t) {
    const int col = n0 + t * 16 + r;
    float bv = ACT ? bias[col] : 0.0f;
#pragma unroll
    for (int i = 0; i < 8; ++i) {
      const int row = m0 + i + 8 * half;
      float v = acc[t][i];
      if (ACT) {
        v += bv;
        v = v / (1.0f + __expf(-v));                  // swish
      }
      C[(size_t)row * N + col] = v;
    }
  }
}

// ---------------------------------------------------- final [512,1] projection
// one wave32 per particle; lane-strided dot + shuffle reduction
__global__ __launch_bounds__(128) void final_dot(
    const float* __restrict__ up, const float* __restrict__ Wf,
    float* __restrict__ out, int M) {
  const int wave = threadIdx.x >> 5;
  const int lane = threadIdx.x & 31;
  const int p = blockIdx.x * 4 + wave;
  if (p >= M) return;
  const float* row = up + (size_t)p * OUTE;
  float s = 0.0f;
#pragma unroll
  for (int d = lane; d < OUTE; d += 32)
    s = __builtin_fmaf(row[d], Wf[d], s);
#pragma unroll
  for (int off = 16; off > 0; off >>= 1)
    s += __shfl_down(s, off, 32);
  if (lane == 0) out[p] = s;
}

// ------------------------------------------------------------------- launcher
extern "C" void kernel_launch(void* const* d_in, const int* in_sizes, int n_in,
                              void* d_out, int out_size, void* d_ws, size_t ws_size,
                              hipStream_t stream) {
  (void)in_sizes; (void)n_in; (void)out_size; (void)ws_size;

  const float* messages = (const float*)d_in[0];
  const float* rbf      = (const float*)d_in[1];
  const int*   idx_i    = (const int*)d_in[2];
  const float* W_rbf    = (const float*)d_in[3];
  const float* W_up     = (const float*)d_in[4];
  const float* W_d0     = (const float*)d_in[5];
  const float* b_d0     = (const float*)d_in[6];
  const float* W_d1     = (const float*)d_in[7];
  const float* b_d1     = (const float*)d_in[8];
  const float* W_final  = (const float*)d_in[9];

  float* summed = (float*)d_ws;                          // [N_PART, EMBED]
  float* buf1   = summed + (size_t)N_PART * EMBED;       // [N_PART, OUTE] (up1/up3)
  float* buf2   = buf1 + (size_t)N_PART * OUTE;          // [N_PART, OUTE] (up2)

  const int n_sum = N_PART * EMBED;
  zero_f32<<<(n_sum + 255) / 256, 256, 0, stream>>>(summed, n_sum);

  edge_gate_scatter<<<N_EDGES / 16, 256, 0, stream>>>(
      messages, rbf, idx_i, W_rbf, summed, N_EDGES);

  gemm_wmma_f32<0><<<N_PART / 16, 256, 0, stream>>>(
      summed, W_up, nullptr, buf1, N_PART, EMBED, OUTE);
  gemm_wmma_f32<1><<<N_PART / 16, 256, 0, stream>>>(
      buf1, W_d0, b_d0, buf2, N_PART, OUTE, OUTE);
  gemm_wmma_f32<1><<<N_PART / 16, 256, 0, stream>>>(
      buf2, W_d1, b_d1, buf1, N_PART, OUTE, OUTE);

  final_dot<<<(N_PART + 3) / 4, 128, 0, stream>>>(
      buf1, W_final, (float*)d_out, N_PART);
}